// MaxPoolKernel_27839978013203
// MI455X (gfx1250) — compile-verified
//
#include <hip/hip_runtime.h>
#include <float.h>
#include <stdint.h>

// CDNA5 / gfx1250, wave32. WMMA f32 16x16x4 builds pairwise-difference tiles
// (dx[m,n] = qx[m]*1 + 1*(-px[n]) -- exact f32, identical to qx-px); box-mask
// bitmap in LDS; bit-scan gather + fmax; float atomic-max merge across N-chunks.

typedef __attribute__((ext_vector_type(2))) float v2f;
typedef __attribute__((ext_vector_type(8))) float v8f;

#define QTILE  16            // queries per workgroup (one WMMA M-tile)
#define NCHUNK 1024          // points per workgroup (grid.y splits N)
#define BLOCK  256           // 8 waves
#define WORDS  (NCHUNK / 32) // 32 mask words per query per chunk

__global__ void __launch_bounds__(BLOCK)
boxmax_init_kernel(float* __restrict__ out, int n) {
    int i = blockIdx.x * BLOCK + threadIdx.x;
    if (i < n) out[i] = -FLT_MAX;  // == jnp.finfo(float32).min
}

__device__ __forceinline__ void atomicMaxF32(float* addr, float v) {
    // Monotonic-encoding trick (no NaNs in this workload).
    if (v >= 0.0f) atomicMax((int*)addr, __float_as_int(v));
    else           atomicMin((unsigned int*)addr, __float_as_uint(v));
}

__device__ __forceinline__ v8f wmma_diff(v2f a, v2f b) {
    v8f cz = {};
    // 8 args: (neg_a, A, neg_b, B, c_mod, C, reuse_a, reuse_b)
    return __builtin_amdgcn_wmma_f32_16x16x4_f32(false, a, false, b,
                                                 (short)0, cz, false, false);
}

__global__ void __launch_bounds__(BLOCK)
boxmax_kernel(const float* __restrict__ values,   // [N, C]
              const float* __restrict__ coords,   // [N, 2]
              const float* __restrict__ qcoords,  // [M, 2]
              float* __restrict__ out,            // [M, C]
              int N, int M, int C) {              // C == 64
    // Bitmap, transposed layout: lmask[widx][row], widx = 32-col group (0..31),
    // row = query row in tile (0..15).  64B-contiguous per widx -> b128 stores.
    __shared__ __align__(16) uint32_t lmask[WORDS * QTILE];  // 2 KB

    const int tid   = threadIdx.x;
    const int lane  = tid & 31;
    const int wave  = tid >> 5;                   // 0..7
    const int m0    = blockIdx.x * QTILE;
    const int nbase = blockIdx.y * NCHUNK;

    // Warm this block's coord chunk (global_prefetch_b8 path on gfx1250).
    {
        const char* p = (const char*)(coords + (size_t)nbase * 2) + (size_t)tid * 32;
        __builtin_prefetch(p, 0, 3);
        if (tid == 0) __builtin_prefetch(qcoords + (size_t)m0 * 2, 0, 3);
    }

    // ---------------- Phase 1: WMMA pairwise diffs -> mask bitmap ----------
    // A (16x4 f32): row m = [q, 1, 0, 0].  Lanes 0-15 carry K=0..1, lanes
    // 16-31 carry K=2..3 (zeros).
    float qx = -1.0e30f, qy = -1.0e30f;
    if (lane < QTILE) {
        int m = m0 + lane;
        if (m < M) { qx = qcoords[2 * m]; qy = qcoords[2 * m + 1]; }
    }
    v2f a_dx; a_dx.x = (lane < QTILE) ? qx   : 0.0f;
              a_dx.y = (lane < QTILE) ? 1.0f : 0.0f;
    v2f a_dy; a_dy.x = (lane < QTILE) ? qy   : 0.0f;
              a_dy.y = (lane < QTILE) ? 1.0f : 0.0f;

    const float T = 0.05f;  // KERNEL_SIZE / 2

    #pragma unroll
    for (int p = 0; p < NCHUNK / (8 * 32); ++p) {         // 4 pairs per wave
        const int n0   = nbase + wave * (NCHUNK / 8) + p * 32;
        const int widx = wave * (NCHUNK / (8 * 32)) + p;  // 0..31
        uint32_t w0[8], w1[8];                            // constant-indexed only

        #pragma unroll
        for (int half = 0; half < 2; ++half) {
            const int nt = n0 + half * 16;
            int n = nt + (lane & 15);
            float px = 1.0e30f, py = 1.0e30f;
            if (n < N) { px = coords[2 * n]; py = coords[2 * n + 1]; }
            // B (4x16 f32): col n = [1, -p, 0, 0]^T.  VGPR0: K=0 (lanes0-15),
            // K=2 (lanes16-31); VGPR1: K=1 / K=3.
            v2f b_dx; b_dx.x = (lane < 16) ? 1.0f : 0.0f;
                      b_dx.y = (lane < 16) ? -px  : 0.0f;
            v2f b_dy; b_dy.x = (lane < 16) ? 1.0f : 0.0f;
                      b_dy.y = (lane < 16) ? -py  : 0.0f;

            v8f dx = wmma_diff(a_dx, b_dx);   // dx[m][n] exact f32 q-p
            v8f dy = wmma_diff(a_dy, b_dy);

            // D layout: VGPR r, lane l -> M = r + 8*(l>=16), Ncol = l&15.
            // ballot word r: bits[15:0] = row r, bits[31:16] = row r+8.
            // ballot(p && q) == ballot(p) & ballot(q).
            #pragma unroll
            for (int r = 0; r < 8; ++r) {
                uint32_t bx = __builtin_amdgcn_ballot_w32(__builtin_fabsf(dx[r]) < T);
                uint32_t by = __builtin_amdgcn_ballot_w32(__builtin_fabsf(dy[r]) < T);
                uint32_t bw = bx & by;
                if (half == 0) w0[r] = bw; else w1[r] = bw;
            }
        }

        // Assemble all 16 per-row 32-bit words (wave-uniform, constant
        // indices -> stays in S/VGPRs, no alloca promotion), then lane 0
        // stores 64B contiguously with 4x ds_store_b128.
        uint32_t wr[16];
        #pragma unroll
        for (int r = 0; r < 8; ++r) {
            wr[r]     = (w0[r] & 0xFFFFu) | (w1[r] << 16);          // row r
            wr[r + 8] = (w0[r] >> 16)     | (w1[r] & 0xFFFF0000u);  // row r+8
        }
        if (lane == 0) {
            uint4* dst = (uint4*)&lmask[widx * QTILE];
            dst[0] = make_uint4(wr[0],  wr[1],  wr[2],  wr[3]);
            dst[1] = make_uint4(wr[4],  wr[5],  wr[6],  wr[7]);
            dst[2] = make_uint4(wr[8],  wr[9],  wr[10], wr[11]);
            dst[3] = make_uint4(wr[12], wr[13], wr[14], wr[15]);
        }
    }
    __syncthreads();

    // ---------------- Phase 2: bit-scan gather + channel max ---------------
    const int c    = tid & 63;   // channel (C == 64)
    const int qsub = tid >> 6;   // 0..3 -> 4 queries at a time (uniform/wave)
    #pragma unroll
    for (int qq = 0; qq < QTILE / 4; ++qq) {
        const int row = qq * 4 + qsub;
        const int m   = m0 + row;
        float acc = -FLT_MAX;
        for (int widx = 0; widx < WORDS; ++widx) {
            uint32_t wv = lmask[widx * QTILE + row];  // uniform per wave
            while (wv) {
                int b = __builtin_ctz(wv);
                wv &= wv - 1;
                int n = nbase + widx * 32 + b;
                acc = fmaxf(acc, values[(size_t)n * C + c]);  // coalesced row
            }
        }
        if (m < M) atomicMaxF32(out + (size_t)m * C + c, acc);
    }
}

extern "C" void kernel_launch(void* const* d_in, const int* in_sizes, int n_in,
                              void* d_out, int out_size, void* d_ws, size_t ws_size,
                              hipStream_t stream) {
    const float* values  = (const float*)d_in[0];  // [N, C]
    const float* coords  = (const float*)d_in[1];  // [N, 2]
    const float* qcoords = (const float*)d_in[2];  // [M, 2]
    float* out = (float*)d_out;

    const int N = in_sizes[1] / 2;
    const int M = in_sizes[2] / 2;
    const int C = in_sizes[0] / N;   // == 64 for this problem

    const int total = M * C;
    boxmax_init_kernel<<<(total + BLOCK - 1) / BLOCK, BLOCK, 0, stream>>>(out, total);

    dim3 grid((M + QTILE - 1) / QTILE, (N + NCHUNK - 1) / NCHUNK);
    boxmax_kernel<<<grid, BLOCK, 0, stream>>>(values, coords, qcoords, out, N, M, C);
}